// VanillaVAE_86268713108331
// MI455X (gfx1250) — compile-verified
//
#include <hip/hip_runtime.h>
#include <math.h>

typedef float v2f __attribute__((ext_vector_type(2)));
typedef float v8f __attribute__((ext_vector_type(8)));
typedef unsigned int v4u __attribute__((ext_vector_type(4)));
typedef int v4i __attribute__((ext_vector_type(4)));
typedef int v8i __attribute__((ext_vector_type(8)));

// D = A(16x4,f32) * B(4x16,f32) + C(16x16,f32)  — exact f32 matrix op on CDNA5
__device__ __forceinline__ v8f wmma4(v2f a, v2f b, v8f c) {
  return __builtin_amdgcn_wmma_f32_16x16x4_f32(
      /*neg_a=*/false, a, /*neg_b=*/false, b,
      /*c_mod=*/(short)0, c, /*reuse_a=*/false, /*reuse_b=*/false);
}

#define ACT_NONE  0
#define ACT_LEAKY 1
#define ACT_SELU  2

// C[M,N] = act(A[M,K] @ W[K,N] + bias[N]); one wave per 16x16 tile, K%16==0.
// K loop chunked by 16 with pointer bumps so loads use immediate offsets.
__global__ void gemm_act_kernel(const float* __restrict__ A,
                                const float* __restrict__ W,
                                const float* __restrict__ bias,
                                float* __restrict__ C,
                                int N, int K, int n_tiles, int act) {
  int wave = blockIdx.x * (blockDim.x >> 5) + (threadIdx.x >> 5);
  if (wave >= n_tiles) return;                  // wave-uniform: EXEC full below
  int tiles_n = N >> 4;
  int tm = wave / tiles_n, tn = wave % tiles_n;
  int lane = threadIdx.x & 31;
  int half = lane >> 4;
  int l    = lane & 15;
  const float* ap = A + (size_t)(tm * 16 + l) * K + half * 2;
  int col = tn * 16 + l;
  const float* wp = W + col + (size_t)(half * 2) * N;
  const size_t nstep = (size_t)16 * N;
  v8f acc = {};
  for (int c = 0; c < (K >> 4); ++c) {
#pragma unroll
    for (int u = 0; u < 4; ++u) {
      v2f a; a.x = ap[u * 4];                  a.y = ap[u * 4 + 1];
      v2f b; b.x = wp[(size_t)(u * 4) * N];    b.y = wp[(size_t)(u * 4 + 1) * N];
      acc = wmma4(a, b, acc);
    }
    ap += 16; wp += nstep;
  }
  float bv = bias[col];
#pragma unroll
  for (int v = 0; v < 8; ++v) {
    int row = tm * 16 + v + half * 8;
    float x = acc[v] + bv;
    if (act == ACT_LEAKY) x = (x >= 0.f) ? x : 0.2f * x;
    else if (act == ACT_SELU) {
      const float sc = 1.0507009873554805f, al = 1.6732632423543772f;
      x = (x > 0.f) ? sc * x : sc * al * expm1f(x);
    }
    C[(size_t)row * N + col] = x;
  }
}

// mu = H@Wmu+bmu ; logvar = H@Wlv+blv ; z = mu + eps*exp(0.5*logvar)
// H:[128,256], Wmu/Wlv:[256,64]. Tiles: 8x4 (32 waves), dual WMMA accumulators.
__global__ void muvar_z_kernel(const float* __restrict__ H,
                               const float* __restrict__ Wmu, const float* __restrict__ bmu,
                               const float* __restrict__ Wlv, const float* __restrict__ blv,
                               const float* __restrict__ eps,
                               float* __restrict__ mu_out, float* __restrict__ lv_out,
                               float* __restrict__ z) {
  int wave = blockIdx.x * (blockDim.x >> 5) + (threadIdx.x >> 5);
  if (wave >= 32) return;
  int tm = wave >> 2, tn = wave & 3;
  int lane = threadIdx.x & 31, half = lane >> 4, l = lane & 15;
  const float* hp = H + (size_t)(tm * 16 + l) * 256 + half * 2;
  int col = tn * 16 + l;
  const float* wm = Wmu + col + half * 2 * 64;
  const float* wl = Wlv + col + half * 2 * 64;
  v8f am = {}, av = {};
  for (int c = 0; c < 16; ++c) {               // 256 / 16
#pragma unroll
    for (int u = 0; u < 4; ++u) {
      v2f a;  a.x  = hp[u * 4];            a.y  = hp[u * 4 + 1];
      v2f b1; b1.x = wm[(u * 4) * 64];     b1.y = wm[(u * 4 + 1) * 64];
      v2f b2; b2.x = wl[(u * 4) * 64];     b2.y = wl[(u * 4 + 1) * 64];
      am = wmma4(a, b1, am);
      av = wmma4(a, b2, av);
    }
    hp += 16; wm += 16 * 64; wl += 16 * 64;
  }
  float bm = bmu[col], bl = blv[col];
#pragma unroll
  for (int v = 0; v < 8; ++v) {
    int row = tm * 16 + v + half * 8;
    float m  = am[v] + bm;
    float lv = av[v] + bl;
    mu_out[row * 64 + col] = m;
    lv_out[row * 64 + col] = lv;
    z[row * 64 + col] = m + eps[row * 64 + col] * expf(0.5f * lv);
  }
}

// pts head: [128,512] @ [512,40] (padded to 48 cols), tanh*12+14, scatter into
// overlapping-cubic cp layout [128,2,3,4,2].
__global__ void head_pts_kernel(const float* __restrict__ H,
                                const float* __restrict__ Wp,
                                const float* __restrict__ bp,
                                float* __restrict__ cp_out) {
  int wave = blockIdx.x * (blockDim.x >> 5) + (threadIdx.x >> 5);
  if (wave >= 24) return;                       // 8 m-tiles x 3 n-tiles
  int tm = wave / 3, tn = wave % 3;
  int lane = threadIdx.x & 31, half = lane >> 4, l = lane & 15;
  int col = tn * 16 + l;
  bool valid = col < 40;
  const float* hrow = H + (size_t)(tm * 16 + l) * 512 + half * 2;
  v8f acc = {};
#pragma unroll 4
  for (int k = 0; k < 512; k += 4) {
    v2f a; a.x = hrow[k]; a.y = hrow[k + 1];
    v2f b;
    b.x = valid ? Wp[(size_t)(k + half * 2) * 40 + col] : 0.f;
    b.y = valid ? Wp[(size_t)(k + half * 2 + 1) * 40 + col] : 0.f;
    acc = wmma4(a, b, acc);
  }
  if (!valid) return;
  float bb = bp[col];
  int p = col / 20;            // path
  int q = (col % 20) >> 1;     // point index in path (0..9)
  int c = col & 1;             // coord
#pragma unroll
  for (int v = 0; v < 8; ++v) {
    int row = tm * 16 + v + half * 8;            // batch index
    float x = tanhf(acc[v] + bb) * 12.0f + 14.0f; // *(CANVAS/2-MARGIN)+CANVAS/2
    for (int s = 0; s < 3; ++s) {                // q = 3s+k overlap scatter
      int kk = q - 3 * s;
      if (kk >= 0 && kk < 4)
        cp_out[(((size_t)row * 2 + p) * 3 + s) * 8 + kk * 2 + c] = x;
    }
  }
}

// widths = sigmoid(H@wd_w+wd_b)*2+1 ; alphas = sigmoid(H@a_w+a_b). Tiny: VALU.
__global__ void head_scalar_kernel(const float* __restrict__ H,
                                   const float* __restrict__ wd_w, const float* __restrict__ wd_b,
                                   const float* __restrict__ a_w,  const float* __restrict__ a_b,
                                   float* __restrict__ widths, float* __restrict__ alphas) {
  int idx = blockIdx.x * blockDim.x + threadIdx.x;
  if (idx >= 128 * 2 * 2) return;
  int which = idx & 1;           // 0: width, 1: alpha
  int p = (idx >> 1) & 1;
  int b = idx >> 2;
  const float* w = which ? a_w : wd_w;
  float s = which ? a_b[p] : wd_b[p];
  const float* h = H + (size_t)b * 512;
#pragma unroll 8
  for (int k = 0; k < 512; ++k) s += h[k] * w[k * 2 + p];
  float sg = 1.f / (1.f + expf(-s));
  if (which) alphas[b * 2 + p] = sg;
  else       widths[b * 2 + p] = sg * 2.f + 1.f;
}

// Evaluate cubic beziers at 32 midpoints/segment -> curveA[b,p,m] = (x,y,x^2+y^2,1)
__global__ void curve_kernel(const float* __restrict__ cp, float* __restrict__ curveA) {
  int idx = blockIdx.x * blockDim.x + threadIdx.x;
  if (idx >= 128 * 2 * 96) return;
  int m = idx % 96;
  int p = (idx / 96) & 1;
  int b = idx / 192;
  int s = m / 32, j = m % 32;
  float t = (j + 0.5f) / 32.f, u = 1.f - t;
  float b0 = u * u * u, b1 = 3.f * t * u * u, b2 = 3.f * t * t * u, b3 = t * t * t;
  const float* c = cp + (((size_t)b * 2 + p) * 3 + s) * 8;
  float x = b0 * c[0] + b1 * c[2] + b2 * c[4] + b3 * c[6];
  float y = b0 * c[1] + b1 * c[3] + b2 * c[5] + b3 * c[7];
  float* o = curveA + (size_t)idx * 4;
  o[0] = x; o[1] = y; o[2] = x * x + y * y; o[3] = 1.f;
}

// WMMA rasterizer with TDM-staged LDS curve data.
// Block = 8 waves, handles 8 sample tiles of one batch image; per-batch curve
// operand block (2*96*4 floats = 3KB) is DMA'd to LDS once by the Tensor Data
// Mover, then all waves feed WMMA A-operands from LDS (ds_load_b64).
// d^2[pt,samp] via A=(cx,cy,|c|^2,1), B=(-2sx,-2sy,1,|s|^2).
#define CURVE_FLOATS (2 * 96 * 4)   // 768 floats per batch
__global__ void render_kernel(const float* __restrict__ curveA,
                              const float* __restrict__ widths,
                              const float* __restrict__ alphas,
                              float* __restrict__ val) {
  __shared__ float smem[CURVE_FLOATS];
  int b = blockIdx.x / 25;
  int g = blockIdx.x % 25;
  const float* src = curveA + (size_t)b * CURVE_FLOATS;

#if __has_builtin(__builtin_amdgcn_tensor_load_to_lds) && \
    __has_builtin(__builtin_amdgcn_s_wait_tensorcnt)
  if (threadIdx.x < 32) {          // one wave issues the TDM descriptor
    unsigned long long ga = (unsigned long long)(uintptr_t)src;
    unsigned int lds_off = (unsigned int)(uintptr_t)(void*)smem; // addr[31:0] = LDS offset
    // D# group 0: count=1 | lds_addr | global_addr | type=2 (ISA §8.3)
    v4u g0 = { 1u, lds_off, (unsigned int)ga,
               (unsigned int)(ga >> 32) | 0x80000000u };
    // D# group 1: data_size=4B; tensor_dim0=tile_dim0=768 elems; dim1=1;
    //             tensor_dim0_stride=768 (ISA §8.4)
    v8i g1 = { 0x00020000, (int)(768u << 16), (int)(1u << 16),
               (int)(768u << 16), 1, 768, 0, 0 };
    v4i g2 = { 0, 0, 0, 0 };       // <=2D tensor: groups 2/3 unused
    v4i g3 = { 0, 0, 0, 0 };
    v8i g4 = { 0, 0, 0, 0, 0, 0, 0, 0 };  // unused trailing group (clang-23 arity)
    __builtin_amdgcn_tensor_load_to_lds(g0, g1, g2, g3, g4, 0);
    __builtin_amdgcn_s_wait_tensorcnt(0);
  }
#else
  for (int i = threadIdx.x; i < CURVE_FLOATS; i += 256) smem[i] = src[i];
#endif
  __syncthreads();

  int w  = threadIdx.x >> 5;
  int st = g * 8 + w;
  if (st >= 196) return;           // wave-uniform: EXEC full below
  int lane = threadIdx.x & 31, half = lane >> 4, l = lane & 15;
  int n = st * 16 + l;             // sample index [0,3136)
  int yy = n / 56, xx = n % 56;
  float sx = (xx + 0.5f) * 0.5f, sy = (yy + 0.5f) * 0.5f;
  v2f bb;
  if (half == 0) { bb.x = -2.f * sx; bb.y = -2.f * sy; }
  else           { bb.x = 1.f;       bb.y = sx * sx + sy * sy; }
  float ink = 1.f;
#pragma unroll
  for (int p = 0; p < 2; ++p) {
    float mn = 3.4e38f;
#pragma unroll
    for (int t = 0; t < 6; ++t) {
      const float* cptr = smem + (p * 96 + t * 16 + l) * 4 + half * 2;
      v2f a; a.x = cptr[0]; a.y = cptr[1];        // ds_load_b64 from LDS
      v8f zero = {};
      v8f d2 = wmma4(a, bb, zero);
#pragma unroll
      for (int v = 0; v < 8; ++v) mn = fminf(mn, d2[v]);
    }
    mn = fminf(mn, __shfl_xor(mn, 16, 32));  // combine row-halves -> min over 96
    float d  = sqrtf(mn + 1e-12f);
    float wd = widths[b * 2 + p], al = alphas[b * 2 + p];
    float cov = 1.f / (1.f + expf(-(wd * 0.5f - d) * 2.f));  // /TAU, TAU=0.5
    ink *= (1.f - al * cov);
  }
  if (half == 0) val[(size_t)b * 3136 + n] = ink;
}

// 2x2 supersample average, white-bg flip: out = 1 - mean
__global__ void downsample_kernel(const float* __restrict__ val, float* __restrict__ out) {
  int idx = blockIdx.x * blockDim.x + threadIdx.x;
  if (idx >= 128 * 784) return;
  int px = idx % 28, py = (idx / 28) % 28, b = idx / 784;
  const float* vb = val + (size_t)b * 3136;
  int y0 = py * 2, x0 = px * 2;
  float s = vb[y0 * 56 + x0] + vb[y0 * 56 + x0 + 1] +
            vb[(y0 + 1) * 56 + x0] + vb[(y0 + 1) * 56 + x0 + 1];
  out[idx] = 1.f - 0.25f * s;
}

extern "C" void kernel_launch(void* const* d_in, const int* in_sizes, int n_in,
                              void* d_out, int out_size, void* d_ws, size_t ws_size,
                              hipStream_t stream) {
  const float* x    = (const float*)d_in[0];
  const float* eps  = (const float*)d_in[1];
  const float* e_w1 = (const float*)d_in[2];
  const float* e_b1 = (const float*)d_in[3];
  const float* e_w2 = (const float*)d_in[4];
  const float* e_b2 = (const float*)d_in[5];
  const float* w_mu = (const float*)d_in[6];
  const float* b_mu = (const float*)d_in[7];
  const float* w_lv = (const float*)d_in[8];
  const float* b_lv = (const float*)d_in[9];
  const float* d_w1 = (const float*)d_in[10];
  const float* d_b1 = (const float*)d_in[11];
  const float* d_w2 = (const float*)d_in[12];
  const float* d_b2 = (const float*)d_in[13];
  const float* p_w  = (const float*)d_in[14];
  const float* p_b  = (const float*)d_in[15];
  const float* wd_w = (const float*)d_in[16];
  const float* wd_b = (const float*)d_in[17];
  const float* a_w  = (const float*)d_in[18];
  const float* a_b  = (const float*)d_in[19];

  float* out      = (float*)d_out;          // return order: rendered, mu, lv, cp, w, a
  float* rendered = out;                    // 128*784  = 100352
  float* mu_out   = out + 100352;           // 128*64   = 8192
  float* lv_out   = out + 108544;           // 128*64   = 8192
  float* cp_out   = out + 116736;           // 128*48   = 6144
  float* w_out    = out + 122880;           // 128*2    = 256
  float* a_out    = out + 123136;           // 128*2    = 256

  float* ws     = (float*)d_ws;
  float* bufA   = ws;                       // 128*512 max = 65536
  float* bufB   = ws + 65536;               // 128*512 max = 65536
  float* zbuf   = ws + 131072;              // 128*64      = 8192
  float* curveA = ws + 139264;              // 128*2*96*4  = 98304
  float* valbuf = ws + 237568;              // 128*3136    = 401408

  // encoder: h1 = leaky(x @ e_w1 + b1)         [128,256], K=784 -> 128 tiles
  gemm_act_kernel<<<32, 128, 0, stream>>>(x, e_w1, e_b1, bufA, 256, 784, 128, ACT_LEAKY);
  // h2 = leaky(h1 @ e_w2 + b2)                 [128,256], K=256 -> 128 tiles
  gemm_act_kernel<<<32, 128, 0, stream>>>(bufA, e_w2, e_b2, bufB, 256, 256, 128, ACT_LEAKY);
  // mu, logvar, z                              [128,64] -> 32 waves
  muvar_z_kernel<<<8, 128, 0, stream>>>(bufB, w_mu, b_mu, w_lv, b_lv, eps,
                                        mu_out, lv_out, zbuf);
  // decoder: h = selu(z @ d_w1 + b)            [128,512], K=64 -> 256 tiles
  gemm_act_kernel<<<64, 128, 0, stream>>>(zbuf, d_w1, d_b1, bufA, 512, 64, 256, ACT_SELU);
  // h = selu(h @ d_w2 + b)                     [128,512], K=512 -> 256 tiles
  gemm_act_kernel<<<64, 128, 0, stream>>>(bufA, d_w2, d_b2, bufB, 512, 512, 256, ACT_SELU);
  // control points (tanh head, N=40 padded to 48) -> 24 waves
  head_pts_kernel<<<6, 128, 0, stream>>>(bufB, p_w, p_b, cp_out);
  // widths & alphas
  head_scalar_kernel<<<2, 256, 0, stream>>>(bufB, wd_w, wd_b, a_w, a_b, w_out, a_out);
  // bezier sampling -> WMMA-ready (x, y, |c|^2, 1) rows
  curve_kernel<<<96, 256, 0, stream>>>(cp_out, curveA);
  // TDM+WMMA rasterizer: 128 batches x 25 blocks x 8 waves
  render_kernel<<<3200, 256, 0, stream>>>(curveA, w_out, a_out, valbuf);
  // 2x2 downsample + 1-x flip
  downsample_kernel<<<392, 256, 0, stream>>>(valbuf, rendered);
}